// _GRU_35871566856224
// MI455X (gfx1250) — compile-verified
//
#include <hip/hip_runtime.h>

// GRU (SEQ=1, h0=0 every layer) fused pipeline for MI455X / gfx1250.
// h0 == 0 makes the recurrent matmul degenerate: gh = b_hh, so w_hh is unused.
// Each wave32 carries 16 batch rows through 5 layers + FC entirely on-chip,
// using V_WMMA_F32_16X16X4_F32 for all matmuls (exact fp32 path).

#define BATCH   1048576
#define E       32          // embed == hidden
#define G3      96          // 3 * hidden
#define NL      5
#define WSTRIDE 34          // padded row stride (floats) for LDS tiles

typedef __attribute__((ext_vector_type(2))) float v2f;
typedef __attribute__((ext_vector_type(4))) float v4f;
typedef __attribute__((ext_vector_type(8))) float v8f;

// ---- LDS layout (floats) ----
#define WTS_SZ  (G3 * WSTRIDE)        // 3264  : current layer w_ih, padded
#define FCW_SZ  (E * WSTRIDE)         // 1088  : fc_w, padded
#define BIH_SZ  (NL * G3)             // 480
#define BHH_SZ  (NL * G3)             // 480
#define FCB_SZ  (E)                   // 32
#define ACT_PER_WAVE (16 * WSTRIDE)   // 544
#define ACT_SZ  (8 * ACT_PER_WAVE)    // 4352 (8 waves / block)
// total = 9696 floats = 38784 bytes static LDS

__device__ __forceinline__ v8f wmma4(v2f a, v2f b, v8f c) {
    // D = A(16x4) * B(4x16) + C, fp32, RNE
    return __builtin_amdgcn_wmma_f32_16x16x4_f32(
        /*neg_a=*/false, a, /*neg_b=*/false, b,
        /*c_mod=*/(short)0, c, /*reuse_a=*/false, /*reuse_b=*/false);
}

__device__ __forceinline__ float fsig(float x) {
    return __builtin_amdgcn_rcpf(1.0f + __expf(-x));
}
__device__ __forceinline__ float ftanh(float x) {
    // tanh(x) = 1 - 2/(e^{2x}+1); saturates correctly at +/-1
    return 1.0f - 2.0f * __builtin_amdgcn_rcpf(1.0f + __expf(2.0f * x));
}

__global__ __launch_bounds__(256, 1)
void gru_fused(const float* __restrict__ x,
               const float* __restrict__ w_ih,
               const float* __restrict__ b_ih,
               const float* __restrict__ b_hh,
               const float* __restrict__ fc_w,
               const float* __restrict__ fc_b,
               float* __restrict__ out)
{
    __shared__ float s_w[WTS_SZ];
    __shared__ float s_fcw[FCW_SZ];
    __shared__ float s_bih[BIH_SZ];
    __shared__ float s_bhh[BHH_SZ];
    __shared__ float s_fcb[FCB_SZ];
    __shared__ float s_act[ACT_SZ];

    const int tid   = threadIdx.x;
    const int lane  = tid & 31;
    const int wave  = tid >> 5;
    const int ncol  = lane & 15;   // N column (B/C/D) and A row supplied by this lane
    const int khalf = lane >> 4;   // which K pair within a K-step

    // ---- one-time staging: fc_w (padded), biases, fc_b ----
    for (int i = tid; i < E * 8; i += 256) {        // 256 v4f chunks
        const int r = i >> 3, kq = (i & 7) << 2;
        v4f v = *(const v4f*)&fc_w[r * E + kq];
        v2f lo, hi; lo.x = v.x; lo.y = v.y; hi.x = v.z; hi.y = v.w;
        *(v2f*)&s_fcw[r * WSTRIDE + kq]     = lo;
        *(v2f*)&s_fcw[r * WSTRIDE + kq + 2] = hi;
    }
    for (int i = tid; i < BIH_SZ; i += 256) s_bih[i] = b_ih[i];
    for (int i = tid; i < BHH_SZ; i += 256) s_bhh[i] = b_hh[i];
    if (tid < FCB_SZ) s_fcb[tid] = fc_b[tid];

    // ---- stage this wave's 16x32 input tile (coalesced b128 reads) ----
    const long rowbase = ((long)blockIdx.x * 8 + wave) * 16;
    float* actw = &s_act[wave * ACT_PER_WAVE];
    {
        const float* xw = x + (size_t)rowbase * E;
        #pragma unroll
        for (int j = 0; j < 4; ++j) {
            const int g = (j << 7) + (lane << 2);   // 0..511
            const int m = g >> 5, k = g & 31;
            v4f v = *(const v4f*)&xw[m * E + k];
            v2f lo, hi; lo.x = v.x; lo.y = v.y; hi.x = v.z; hi.y = v.w;
            *(v2f*)&actw[m * WSTRIDE + k]     = lo;
            *(v2f*)&actw[m * WSTRIDE + k + 2] = hi;
        }
    }

    const v8f Z = {0.f, 0.f, 0.f, 0.f, 0.f, 0.f, 0.f, 0.f};

    // ---- 5 GRU layers ----
    #pragma unroll 1
    for (int l = 0; l < NL; ++l) {
        // stage layer weights into padded LDS tile (reused by all 8 waves)
        __syncthreads();            // previous layer's weight reads are done
        {
            const float* Wl = w_ih + (size_t)l * G3 * E;
            for (int i = tid; i < G3 * 8; i += 256) {     // 768 v4f chunks
                const int r = i >> 3, kq = (i & 7) << 2;
                v4f v = *(const v4f*)&Wl[r * E + kq];
                v2f lo, hi; lo.x = v.x; lo.y = v.y; hi.x = v.z; hi.y = v.w;
                *(v2f*)&s_w[r * WSTRIDE + kq]     = lo;
                *(v2f*)&s_w[r * WSTRIDE + kq + 2] = hi;
            }
        }
        __syncthreads();

        // gi = X(16x32) @ W^T -> six 16x16 accumulators (r0 r1 z0 z1 n0 n1)
        v8f a0 = Z, a1 = Z, a2 = Z, a3 = Z, a4 = Z, a5 = Z;
        #pragma unroll
        for (int kk = 0; kk < 8; ++kk) {
            const int kb = (kk << 2) + (khalf << 1);
            v2f av = *(const v2f*)&actw[ncol * WSTRIDE + kb];
            v2f b0 = *(const v2f*)&s_w[(ncol     ) * WSTRIDE + kb];
            v2f b1 = *(const v2f*)&s_w[(ncol + 16) * WSTRIDE + kb];
            v2f b2 = *(const v2f*)&s_w[(ncol + 32) * WSTRIDE + kb];
            v2f b3 = *(const v2f*)&s_w[(ncol + 48) * WSTRIDE + kb];
            v2f b4 = *(const v2f*)&s_w[(ncol + 64) * WSTRIDE + kb];
            v2f b5 = *(const v2f*)&s_w[(ncol + 80) * WSTRIDE + kb];
            a0 = wmma4(av, b0, a0);
            a1 = wmma4(av, b1, a1);
            a2 = wmma4(av, b2, a2);
            a3 = wmma4(av, b3, a3);
            a4 = wmma4(av, b4, a4);
            a5 = wmma4(av, b5, a5);
        }

        // gates (elementwise in C/D layout; per-column bias scalars)
        const float* bi = &s_bih[l * G3];
        const float* bh = &s_bhh[l * G3];
        const float bir0 = bi[ncol],      bhr0 = bh[ncol];
        const float bir1 = bi[ncol + 16], bhr1 = bh[ncol + 16];
        const float biz0 = bi[ncol + 32], bhz0 = bh[ncol + 32];
        const float biz1 = bi[ncol + 48], bhz1 = bh[ncol + 48];
        const float bin0 = bi[ncol + 64], bhn0 = bh[ncol + 64];
        const float bin1 = bi[ncol + 80], bhn1 = bh[ncol + 80];

        v8f h0, h1;
        #pragma unroll
        for (int i = 0; i < 8; ++i) {
            const float r0 = fsig(a0[i] + bir0 + bhr0);
            const float r1 = fsig(a1[i] + bir1 + bhr1);
            const float z0 = fsig(a2[i] + biz0 + bhz0);
            const float z1 = fsig(a3[i] + biz1 + bhz1);
            const float n0 = ftanh(a4[i] + bin0 + r0 * bhn0);
            const float n1 = ftanh(a5[i] + bin1 + r1 * bhn1);
            h0[i] = (1.0f - z0) * n0;   // h_new = (1-z)*n + z*0
            h1[i] = (1.0f - z1) * n1;
        }

        // write h back to the wave-private activation tile (D-layout scatter);
        // LDS ops are in-order per wave, so next layer's A-loads see this.
        #pragma unroll
        for (int v = 0; v < 8; ++v) {
            const int m = v + (khalf << 3);
            actw[m * WSTRIDE + ncol]      = h0[v];
            actw[m * WSTRIDE + ncol + 16] = h1[v];
        }
        asm volatile("" ::: "memory");   // keep store->load ordering explicit
    }

    // ---- FC: out = h @ fc_w^T + fc_b ----
    v8f o0 = Z, o1 = Z;
    #pragma unroll
    for (int kk = 0; kk < 8; ++kk) {
        const int kb = (kk << 2) + (khalf << 1);
        v2f av = *(const v2f*)&actw[ncol * WSTRIDE + kb];
        v2f b0 = *(const v2f*)&s_fcw[(ncol     ) * WSTRIDE + kb];
        v2f b1 = *(const v2f*)&s_fcw[(ncol + 16) * WSTRIDE + kb];
        o0 = wmma4(av, b0, o0);
        o1 = wmma4(av, b1, o1);
    }
    const float c0 = s_fcb[ncol], c1 = s_fcb[ncol + 16];
    float* op = out + (size_t)rowbase * E;
    #pragma unroll
    for (int v = 0; v < 8; ++v) {
        const int m = v + (khalf << 3);
        op[m * E + ncol]      = o0[v] + c0;
        op[m * E + ncol + 16] = o1[v] + c1;
    }
}

extern "C" void kernel_launch(void* const* d_in, const int* in_sizes, int n_in,
                              void* d_out, int out_size, void* d_ws, size_t ws_size,
                              hipStream_t stream) {
    const float* x    = (const float*)d_in[0];
    const float* w_ih = (const float*)d_in[1];
    // d_in[2] = w_hh : provably unused (h0 == 0 for every layer => h @ w_hh^T == 0)
    const float* b_ih = (const float*)d_in[3];
    const float* b_hh = (const float*)d_in[4];
    const float* fc_w = (const float*)d_in[5];
    const float* fc_b = (const float*)d_in[6];
    float* out = (float*)d_out;

    dim3 grid(BATCH / 128);   // 8 waves/block, 16 rows/wave
    dim3 block(256);
    gru_fused<<<grid, block, 0, stream>>>(x, w_ih, b_ih, b_hh, fc_w, fc_b, out);
}